// LiquidBaseline_10952166605234
// MI455X (gfx1250) — compile-verified
//
#include <hip/hip_runtime.h>
#include <math.h>
#include <stdint.h>

// ---------------------------------------------------------------------------
// LiquidBaseline on MI455X (gfx1250).
//   1) lb_init:   H1[m][n] = b1[n]                         (256x512 acc in ws)
//   2) lb_gemm1:  H1 += x[256 x 268203] @ W1[268203 x 512]
//                 - V_WMMA_F32_16X16X4_F32, 128 acc VGPRs/wave (16 M-tiles)
//                 - X staged into LDS by the Tensor Data Mover (double-buffered,
//                   HW padding gives the bank-conflict-free stride-10 layout)
//                 - W1 streamed nontemporal (used exactly once)
//                 - K-split over blocks, f32 global atomics into L2-resident acc
//                 - bounds checks only in the single tail K-chunk
//   3) lb_u:      U[m][h] = relu(H1[m]) @ Wi               (256x28 in ws)
//   4) lb_rec:    sequential LTC scan over T=64 + sigmoid head -> d_out[8]
// ---------------------------------------------------------------------------

typedef float v2f __attribute__((ext_vector_type(2)));
typedef float v8f __attribute__((ext_vector_type(8)));
typedef unsigned int u4i __attribute__((ext_vector_type(4)));
typedef int s8i __attribute__((ext_vector_type(8)));
typedef int s4i __attribute__((ext_vector_type(4)));

#define K_FEAT 268203   // 3*299*299
#define D1     512
#define MROWS  256      // B*T
#define HDIM   28
#define TSTEPS 64
#define BATCH  4

#define KCHUNK     2096  // per-block K slice, multiple of 8; 128*2096 >= K_FEAT
#define NKCHUNKS   128
#define LDS_STRIDE 10    // 8 k-values + 2 pad floats (8B aligned, conflict-free)

// -------------------------------------------------- init: H1 = bias (broadcast)
__global__ void lb_init_kernel(float* __restrict__ H1, const float* __restrict__ b1) {
  int idx = blockIdx.x * blockDim.x + threadIdx.x;   // 256*512 = 131072 threads
  H1[idx] = b1[idx & (D1 - 1)];
}

// -------------------------------------------------- TDM: X[0..255][k0..k0+7] -> LDS
// 2D tile, data_size=4B, tile = 8 x 256, row stride K_FEAT elements.
// pad_enable: every 8 DWORDs insert 2 DWORDs -> LDS row stride 10 floats.
// tensor_dim0 = K_FEAT - k0 so the final K tail is zero-filled by OOB reads.
__device__ __forceinline__ void tdm_load_slice(const float* __restrict__ X,
                                               long k0, float* ldsdst) {
  const uint64_t ga  = (uint64_t)(uintptr_t)(X + k0);
  const uint32_t lda = (uint32_t)(uintptr_t)ldsdst;     // low 32 bits = LDS offset
  const uint32_t rem = (uint32_t)(K_FEAT - k0);

  u4i g0;
  g0.x = 1u;                                            // count=1 (valid), load, no gather
  g0.y = lda;                                           // lds_addr
  g0.z = (uint32_t)ga;                                  // global_addr[31:0]
  g0.w = (uint32_t)(ga >> 32) | (2u << 30);             // global_addr[56:32] | type=2

  // group1 bitfields (ISA 8.4):
  //  d0: wg_mask=0 | data_size=2(4B)<<16 | pad_en<<20 | pad_interval=2(8DW)<<22
  //      | pad_amount=1(2DW)<<25
  s8i g1;
  g1[0] = (int)((2u << 16) | (1u << 20) | (2u << 22) | (1u << 25));
  g1[1] = (int)((rem & 0xFFFFu) << 16);                 // tensor_dim0[15:0] @ bits63:48
  g1[2] = (int)((rem >> 16) | ((uint32_t)MROWS << 16)); // dim0[31:16] | tensor_dim1[15:0]
  g1[3] = (int)(8u << 16);                              // dim1[31:16]=0 | tile_dim0=8
  g1[4] = (int)(MROWS);                                 // tile_dim1=256 | tile_dim2=0
  g1[5] = (int)(K_FEAT);                                // tensor_dim0_stride[31:0]
  g1[6] = 0;                                            // stride[47:32]=0 | dim1_stride lo
  g1[7] = 0;

  s4i z4 = {0, 0, 0, 0};                                // 2D: groups 2/3 unused
  s8i z8 = {0, 0, 0, 0, 0, 0, 0, 0};
  __builtin_amdgcn_tensor_load_to_lds(g0, g1, z4, z4, z8, 0);
}

// -------------------------------------------------- K-chunk runner
// CHECK=false: all k in [kbeg, kbeg+KCHUNK) are in-bounds -> unguarded loads.
// CHECK=true : tail chunk, per-lane guards on W1 (TDM zero-fills the X tail).
template <bool CHECK>
__device__ __forceinline__ void gemm_chunk(const float* __restrict__ X,
                                           const float* __restrict__ W1,
                                           float* lds0, float* lds1,
                                           v8f acc[16], long kbeg,
                                           int wave, int l16, int khalf, int col) {
  const int nit = KCHUNK / 8;
  if (wave == 0) tdm_load_slice(X, kbeg, lds0);         // prologue DMA

  for (int it = 0; it < nit; ++it) {
    const long k0 = kbeg + (long)it * 8;

    if (wave == 0) {
      if (it + 1 < nit) {
        tdm_load_slice(X, k0 + 8, (it & 1) ? lds0 : lds1);  // prefetch next slice
        __builtin_amdgcn_s_wait_tensorcnt(1);               // current slice landed
      } else {
        __builtin_amdgcn_s_wait_tensorcnt(0);
      }
    }
    __syncthreads();                                    // slice visible to all waves

    const float* buf = (it & 1) ? lds1 : lds0;
    const float* wp  = W1 + (k0 + khalf) * D1 + col;    // per-lane W1 base
#pragma unroll
    for (int half = 0; half < 2; ++half) {              // two k=4 WMMA steps
      v2f bfrag;
      if (CHECK) {
        const long kA = k0 + half * 4 + khalf;
        bfrag.x = (kA     < K_FEAT) ? __builtin_nontemporal_load(&wp[half * 4 * D1])      : 0.0f;
        bfrag.y = (kA + 1 < K_FEAT) ? __builtin_nontemporal_load(&wp[half * 4 * D1 + D1]) : 0.0f;
      } else {
        bfrag.x = __builtin_nontemporal_load(&wp[half * 4 * D1]);
        bfrag.y = __builtin_nontemporal_load(&wp[half * 4 * D1 + D1]);
      }
#pragma unroll
      for (int mt = 0; mt < 16; ++mt) {
        const int row = mt * 16 + l16;
        const v2f afrag = *(const v2f*)&buf[row * LDS_STRIDE + half * 4 + khalf];
        acc[mt] = __builtin_amdgcn_wmma_f32_16x16x4_f32(
            false, afrag, false, bfrag, (short)0, acc[mt], false, false);
      }
    }
    __syncthreads();                                    // reads done before next DMA
  }
}

// -------------------------------------------------- big GEMM, f32 WMMA
// grid = (2, NKCHUNKS), block = 512 (16 waves). Wave w owns 16 columns
// (blockIdx.x*256 + w*16) and all 256 rows as 16 WMMA accumulator tiles.
__global__ __launch_bounds__(512)
void lb_gemm1_kernel(const float* __restrict__ X,
                     const float* __restrict__ W1,
                     float* __restrict__ H1) {
  __shared__ float lds[2][MROWS * LDS_STRIDE];          // double-buffered X slices

  const int tid   = threadIdx.x;
  const int wave  = tid >> 5;
  const int lane  = tid & 31;
  const int l16   = lane & 15;
  const int khalf = (lane >> 4) << 1;                   // 0 (lanes 0-15) / 2 (16-31)
  const int col   = blockIdx.x * 256 + wave * 16 + l16;
  const long kbeg = (long)blockIdx.y * KCHUNK;

  v8f acc[16] = {};                                     // 128 VGPRs of accumulators

  if (kbeg + KCHUNK <= K_FEAT) {                        // 127 of 128 chunks
    gemm_chunk<false>(X, W1, &lds[0][0], &lds[1][0], acc, kbeg, wave, l16, khalf, col);
  } else {                                              // single tail chunk
    gemm_chunk<true>(X, W1, &lds[0][0], &lds[1][0], acc, kbeg, wave, l16, khalf, col);
  }

  // Reduce partial tiles into the global accumulator (native f32 atomic add).
  const int mbase = (lane >> 4) << 3;                   // +8 rows for upper half-wave
#pragma unroll
  for (int mt = 0; mt < 16; ++mt) {
#pragma unroll
    for (int r = 0; r < 8; ++r) {
      const int m = mt * 16 + mbase + r;
      unsafeAtomicAdd(&H1[m * D1 + col], acc[mt][r]);
    }
  }
}

// -------------------------------------------------- U = relu(H1) @ Wi  (tiny)
__global__ void lb_u_kernel(const float* __restrict__ H1, const float* __restrict__ Wi,
                            float* __restrict__ U) {
  const int m = blockIdx.x;                             // 256
  const int h = threadIdx.x;                            // 32 (28 active)
  if (h >= HDIM) return;
  float s = 0.0f;
  for (int d = 0; d < D1; ++d) {
    s += fmaxf(H1[m * D1 + d], 0.0f) * Wi[d * HDIM + h];
  }
  U[m * HDIM + h] = s;
}

// -------------------------------------------------- LTC scan + sigmoid head
__global__ void lb_rec_kernel(const float* __restrict__ U,  const float* __restrict__ Wr,
                              const float* __restrict__ br, const float* __restrict__ Av,
                              const float* __restrict__ tau, const float* __restrict__ W2,
                              const float* __restrict__ b2, float* __restrict__ out) {
  __shared__ float hs[BATCH][32];
  const int tid = threadIdx.x;                          // 128
  const int b   = tid >> 5;
  const int j   = tid & 31;
  hs[b][j] = 0.0f;
  __syncthreads();

  const float rtau = (j < HDIM) ? (1.0f / tau[j]) : 0.0f;
  const float brj  = (j < HDIM) ? br[j] : 0.0f;
  const float Aj   = (j < HDIM) ? Av[j] : 0.0f;

  for (int t = 0; t < TSTEPS; ++t) {
    float hn = 0.0f;
    if (j < HDIM) {
      float dot = 0.0f;
#pragma unroll
      for (int i = 0; i < HDIM; ++i) dot += hs[b][i] * Wr[i * HDIM + j];
      const float z  = U[(b * TSTEPS + t) * HDIM + j] + dot + brj;
      const float ft = 1.0f / (1.0f + expf(-z));
      hn = (hs[b][j] + ft * Aj) / (1.0f + rtau + ft);   // dt = 1
    }
    __syncthreads();
    if (j < HDIM) hs[b][j] = hn;
    __syncthreads();
  }

  if (j < 2) {                                          // out = sigmoid(relu(h_T)@W2+b2)
    float s = b2[j];
    for (int i = 0; i < HDIM; ++i) s += fmaxf(hs[b][i], 0.0f) * W2[i * 2 + j];
    out[b * 2 + j] = 1.0f / (1.0f + expf(-s));
  }
}

// ---------------------------------------------------------------------------
extern "C" void kernel_launch(void* const* d_in, const int* in_sizes, int n_in,
                              void* d_out, int out_size, void* d_ws, size_t ws_size,
                              hipStream_t stream) {
  const float* X   = (const float*)d_in[0];
  const float* W1  = (const float*)d_in[1];
  const float* b1  = (const float*)d_in[2];
  const float* Wi  = (const float*)d_in[3];
  const float* Wr  = (const float*)d_in[4];
  const float* br  = (const float*)d_in[5];
  const float* Av  = (const float*)d_in[6];
  const float* tau = (const float*)d_in[7];
  const float* W2  = (const float*)d_in[8];
  const float* b2  = (const float*)d_in[9];

  float* H1 = (float*)d_ws;                                       // 256*512 f32
  float* U  = (float*)((char*)d_ws + MROWS * D1 * sizeof(float)); // 256*28  f32
  float* out = (float*)d_out;

  lb_init_kernel<<<256, 512, 0, stream>>>(H1, b1);
  lb_gemm1_kernel<<<dim3(2, NKCHUNKS), 512, 0, stream>>>(X, W1, H1);
  lb_u_kernel<<<MROWS, 32, 0, stream>>>(H1, Wi, U);
  lb_rec_kernel<<<1, 128, 0, stream>>>(U, Wr, br, Av, tau, W2, b2, out);
}